// AttentiveQuantizer_17815524344474
// MI455X (gfx1250) — compile-verified
//
#include <hip/hip_runtime.h>

// ---------------------------------------------------------------------------
// AttentiveQuantizer forward, fused for MI455X (gfx1250, wave32, WMMA, async
// LDS DMA).  HBM-bound on the 256 MB logit stream -> full-f32 WMMA is free.
//
//   q = x@wq^T ; k = cb@wk^T ; v = cb@wv^T         (proj_kernel x3)
//   logit = q@k^T/16  (NT b128 stream, written once)
//   soft_v = softmax(logit)@v  (online softmax, fused, double-buffered
//            k/v chunks staged with GLOBAL_LOAD_ASYNC_TO_LDS_B128)
//   quant = v[argmax] straight-through ; code = argmax & 0xFF
// ---------------------------------------------------------------------------

typedef float v2f __attribute__((ext_vector_type(2)));
typedef float v4f __attribute__((ext_vector_type(4)));
typedef float v8f __attribute__((ext_vector_type(8)));

__device__ __forceinline__ v8f wmma4(v2f a, v2f b, v8f c) {
  // D(16x16 f32) = A(16x4 f32) * B(4x16 f32) + C
  return __builtin_amdgcn_wmma_f32_16x16x4_f32(
      false, a, false, b, (short)0, c, false, false);
}

// gfx1250 async DMA: global -> LDS, tracked by ASYNCcnt (no VGPR bounce).
__device__ __forceinline__ void async_b128(unsigned lds_addr,
                                           unsigned long long gaddr) {
  asm volatile("global_load_async_to_lds_b128 %0, %1, off"
               :: "v"(lds_addr), "v"(gaddr) : "memory");
}
__device__ __forceinline__ void wait_async0() {
  asm volatile("s_wait_asynccnt 0x0" ::: "memory");
}
// Low 32 bits of a flat LDS pointer == LDS byte offset (ISA aperture rule).
__device__ __forceinline__ unsigned lds_off(const void* p) {
  return (unsigned)(unsigned long long)p;
}

// ---------------------------------------------------------------------------
// Projection GEMM: out[m][j] = sum_c in[m][c] * W[j][c]   (x @ W^T)
// latent_mode: in[m][c] = latent[(m>>10)*262144 + c*1024 + (m&1023)]
// ---------------------------------------------------------------------------
__global__ __launch_bounds__(128) void proj_kernel(
    const float* __restrict__ in, const float* __restrict__ W,
    float* __restrict__ out, float* __restrict__ out2, int latent_mode) {
  __shared__ float As[32][36];   // stride 36 = 4*9 -> conflict-free frag reads
  __shared__ float Bs[32][36];

  const int tid  = threadIdx.x;
  const int wave = tid >> 5;
  const int lane = tid & 31;
  const int ml   = lane & 15;
  const int hi   = lane >> 4;
  const int m0   = blockIdx.x * 32;
  const int j0   = blockIdx.y * 32;
  const int mt   = (wave >> 1) * 16;
  const int jt   = (wave & 1) * 16;

  v8f acc;
  for (int g = 0; g < 8; ++g) acc[g] = 0.0f;

  for (int c0 = 0; c0 < 256; c0 += 32) {
    if (latent_mode) {
      const int n   = m0 >> 10;
      const int hw0 = m0 & 1023;
      const float* base = in + (size_t)n * 262144 + hw0;
      for (int i = tid; i < 32 * 32; i += 128) {
        int mm = i & 31, cc = i >> 5;                 // coalesced along hw
        As[mm][cc] = base[(size_t)(c0 + cc) * 1024 + mm];
      }
    } else {
      for (int i = tid; i < 32 * 32; i += 128) {
        int cc = i & 31, mm = i >> 5;                 // coalesced along c
        As[mm][cc] = in[(size_t)(m0 + mm) * 256 + c0 + cc];
      }
    }
    for (int i = tid; i < 32 * 32; i += 128) {
      int cc = i & 31, jj = i >> 5;
      Bs[jj][cc] = W[(size_t)(j0 + jj) * 256 + c0 + cc];
    }
    __syncthreads();

    for (int k = 0; k < 32; k += 4) {
      const int cA = k + 2 * hi;
      v2f a = *(const v2f*)&As[mt + ml][cA];   // A: [M=ml][K=g+2*hi]
      v2f b = *(const v2f*)&Bs[jt + ml][cA];   // B: [K=g+2*hi][N=ml]
      acc = wmma4(a, b, acc);
    }
    __syncthreads();
  }

  const int gm = m0 + mt + 8 * hi;
  const int gj = j0 + jt + ml;
  for (int g = 0; g < 8; ++g)
    out[(size_t)(gm + g) * 256 + gj] = acc[g];
  if (out2 != nullptr) {
    for (int g = 0; g < 8; ++g)
      out2[(size_t)(gm + g) * 256 + gj] = acc[g];
  }
}

// ---------------------------------------------------------------------------
// Fused attention: block = 64 rows, 8 waves; 4096 codes in 128 chunks of 32.
// k/v chunks double-buffered in LDS via async DMA.
// ---------------------------------------------------------------------------
#define KC 32

__global__ __launch_bounds__(256) void fused_attn_kernel(
    const float* __restrict__ qbuf, const float* __restrict__ kbuf,
    const float* __restrict__ vbuf, float* __restrict__ quant,
    float* __restrict__ code, float* __restrict__ logit) {
  __shared__ float Qs[64][260];       // stride 260 ≡ 4 (mod 64)
  __shared__ float Kb[2][KC][260];    // double-buffered k chunk [j][c]
  __shared__ float Vb[2][KC][264];    // double-buffered v chunk [j][c]
  __shared__ float Ss[64][36];        // scaled logits of current chunk
  __shared__ float Ps[64][36];        // exp(S - rowmax)
  __shared__ float rowmax[64], rowsum[64], rowscale[64], rowinvl[64];
  __shared__ int   rowarg[64];

  const int tid  = threadIdx.x;
  const int wave = tid >> 5;
  const int lane = tid & 31;
  const int ml   = lane & 15;
  const int hi   = lane >> 4;
  const int m0   = blockIdx.x * 64;

  // Load the 64x256 q slab once (coalesced).
  for (int i = tid; i < 64 * 256; i += 256) {
    int r = i >> 8, cc = i & 255;
    Qs[r][cc] = qbuf[(size_t)(m0 + r) * 256 + cc];
  }
  if (tid < 64) {
    rowmax[tid] = -3.402823466e38f;
    rowsum[tid] = 0.0f;
    rowarg[tid] = 0;
  }

  v8f acc[4][2];
  for (int rt = 0; rt < 4; ++rt)
    for (int ct = 0; ct < 2; ++ct)
      for (int g = 0; g < 8; ++g) acc[rt][ct][g] = 0.0f;

  const int smt = (wave >> 1) * 16;   // S-tile row base
  const int sjt = (wave & 1) * 16;    // S-tile col base
  const int cb0 = wave * 32;          // wave's 32 output channels

  // Stage chunk 0 into buffer 0 with async DMA (16B per lane per op).
  for (int i = tid; i < KC * 64; i += 256) {
    int r = i >> 6, c4 = (i & 63) << 2;
    async_b128(lds_off(&Kb[0][r][c4]),
               (unsigned long long)(kbuf + (size_t)r * 256 + c4));
    async_b128(lds_off(&Vb[0][r][c4]),
               (unsigned long long)(vbuf + (size_t)r * 256 + c4));
  }
  wait_async0();
  __syncthreads();

  for (int ch = 0; ch < 128; ++ch) {
    const int j0  = ch * KC;
    const int cur = ch & 1;

    // Kick off DMA for the next chunk into the other buffer.
    if (ch + 1 < 128) {
      const size_t gb = (size_t)(j0 + KC) * 256;
      for (int i = tid; i < KC * 64; i += 256) {
        int r = i >> 6, c4 = (i & 63) << 2;
        async_b128(lds_off(&Kb[cur ^ 1][r][c4]),
                   (unsigned long long)(kbuf + gb + (size_t)r * 256 + c4));
        async_b128(lds_off(&Vb[cur ^ 1][r][c4]),
                   (unsigned long long)(vbuf + gb + (size_t)r * 256 + c4));
      }
    }

    // ---- S = q @ k^T, 16x16 tile per wave, K=256 -> 64 WMMAs ----
    v8f s;
    for (int g = 0; g < 8; ++g) s[g] = 0.0f;
    for (int c = 0; c < 256; c += 4) {
      const int cA = c + 2 * hi;
      v2f a = *(const v2f*)&Qs[smt + ml][cA];
      v2f b = *(const v2f*)&Kb[cur][sjt + ml][cA];
      s = wmma4(a, b, s);
    }
    for (int g = 0; g < 8; ++g)
      Ss[smt + g + 8 * hi][sjt + ml] = s[g] * 0.0625f;   // /sqrt(256)
    __syncthreads();

    // ---- stream logits (NT b128: 4 threads per 128B row) ----
    {
      const int r  = tid >> 2;
      const int cg = (tid & 3) * 8;
      v4f o0 = *(const v4f*)&Ss[r][cg];
      v4f o1 = *(const v4f*)&Ss[r][cg + 4];
      float* dst = logit + (size_t)(m0 + r) * 4096 + j0 + cg;
      __builtin_nontemporal_store(o0, (v4f*)dst);
      __builtin_nontemporal_store(o1, (v4f*)(dst + 4));
    }
    // ---- online softmax stats, one thread per row (reads Ss, writes Ps) ----
    if (tid < 64) {
      const float mo = rowmax[tid];
      float mx = -3.402823466e38f;
      int   ag = 0;
      for (int j = 0; j < KC; ++j) {
        const float sv = Ss[tid][j];
        if (sv > mx) { mx = sv; ag = j; }     // first max within chunk
      }
      const float mn = (mo > mx) ? mo : mx;
      if (mx > mo) rowarg[tid] = j0 + ag;     // strict > keeps earliest
      const float sc = __expf(mo - mn);
      float psum = 0.0f;
      for (int j = 0; j < KC; ++j) {
        const float p = __expf(Ss[tid][j] - mn);
        Ps[tid][j] = p;
        psum += p;
      }
      rowsum[tid]   = rowsum[tid] * sc + psum;
      rowmax[tid]   = mn;
      rowscale[tid] = sc;
    }
    __syncthreads();

    // ---- acc = acc*rescale + P @ v_chunk  (4x2 tiles x 8 WMMAs) ----
    for (int rt = 0; rt < 4; ++rt) {
      const int r0 = rt * 16 + 8 * hi;
      float scv[8];
      for (int g = 0; g < 8; ++g) scv[g] = rowscale[r0 + g];
      for (int ct = 0; ct < 2; ++ct) {
        v8f a8 = acc[rt][ct];
        for (int g = 0; g < 8; ++g) a8[g] *= scv[g];
        const int colb = cb0 + ct * 16 + ml;
        for (int k = 0; k < KC; k += 4) {
          const int kA = k + 2 * hi;
          v2f a = *(const v2f*)&Ps[rt * 16 + ml][kA];
          v2f b;
          b.x = Vb[cur][kA][colb];
          b.y = Vb[cur][kA + 1][colb];
          a8 = wmma4(a, b, a8);
        }
        acc[rt][ct] = a8;
      }
    }

    wait_async0();      // next chunk's DMA landed
    __syncthreads();    // everyone done with this chunk's buffers
  }

  // ---- epilogue ----
  if (tid < 64) {
    rowinvl[tid] = 1.0f / rowsum[tid];
    code[m0 + tid] = (float)(rowarg[tid] & 0xFF);   // .byte() wrap
  }
  __syncthreads();

  const int n   = m0 >> 10;     // 64-row slab never crosses an image
  const int hw0 = m0 & 1023;
  for (int rt = 0; rt < 4; ++rt) {
    const int r0 = rt * 16 + 8 * hi;
    for (int ct = 0; ct < 2; ++ct) {
      const int col = cb0 + ct * 16 + ml;
      float ov[8];
      for (int g = 0; g < 8; ++g) {
        const int r = r0 + g;
        const float soft = acc[rt][ct][g] * rowinvl[r];
        const float hard = vbuf[(size_t)rowarg[r] * 256 + col];
        ov[g] = (hard - soft) + soft;       // straight-through forward
      }
      // NCHW: for a fixed channel, rows are contiguous along hw.
      float* dst = quant + (size_t)n * 262144 + (size_t)col * 1024 + hw0 + r0;
      v4f o0 = {ov[0], ov[1], ov[2], ov[3]};
      v4f o1 = {ov[4], ov[5], ov[6], ov[7]};
      __builtin_nontemporal_store(o0, (v4f*)dst);
      __builtin_nontemporal_store(o1, (v4f*)(dst + 4));
    }
  }
}

// ---------------------------------------------------------------------------
extern "C" void kernel_launch(void* const* d_in, const int* in_sizes, int n_in,
                              void* d_out, int out_size, void* d_ws, size_t ws_size,
                              hipStream_t stream) {
  (void)in_sizes; (void)n_in; (void)out_size; (void)ws_size;
  const float* latent   = (const float*)d_in[0];   // [16,256,32,32]
  const float* codebook = (const float*)d_in[1];   // [4096,256]
  const float* wq       = (const float*)d_in[2];   // [256,256]
  const float* wk       = (const float*)d_in[3];
  const float* wv       = (const float*)d_in[4];
  // d_in[5] = temperature (deterministic path: unused)

  float* out   = (float*)d_out;
  float* quant = out;                // 16*256*32*32 = 4194304
  float* code  = out + 4194304;      // 16384
  float* logit = out + 4210688;      // 16384*4096 = 67108864
  float* vout  = out + 71319552;     // 4096*256

  float* qbuf = (float*)d_ws;        // 16384*256
  float* kbuf = qbuf + 4194304;      // 4096*256
  float* vbuf = kbuf + 1048576;      // 4096*256

  proj_kernel<<<dim3(512, 8), dim3(128), 0, stream>>>(latent,   wq, qbuf, nullptr, 1);
  proj_kernel<<<dim3(128, 8), dim3(128), 0, stream>>>(codebook, wk, kbuf, nullptr, 0);
  proj_kernel<<<dim3(128, 8), dim3(128), 0, stream>>>(codebook, wv, vbuf, vout,    0);
  fused_attn_kernel<<<dim3(256), dim3(256), 0, stream>>>(qbuf, kbuf, vbuf,
                                                         quant, code, logit);
}